// OHEMLoss_41446434406852
// MI455X (gfx1250) — compile-verified
//
#include <hip/hip_runtime.h>
#include <stdint.h>

#define BATCH 8
#define NPIX  (1u << 20)          // 1024*1024 per row
#define KALL  524288              // N * KEEP_RATIO
#define TPB   256
#define NPB   512                 // pass1 blocks per row
#define HPB   256                 // hist pass blocks per row
#define SPB   512                 // sum pass blocks per row

typedef float v2f __attribute__((ext_vector_type(2)));
typedef float v8f __attribute__((ext_vector_type(8)));

// ---------------------------------------------------------------- init ----
__global__ void ohem_init_kernel(uint32_t* __restrict__ hist, int* __restrict__ nstats) {
    int i = blockIdx.x * blockDim.x + threadIdx.x;
    if (i < 3 * BATCH * 2048) hist[i] = 0u;
    if (i < 2 * BATCH) nstats[i] = 0;
}

// ---------------------------------------------------------------- pass1 ---
// Compute BCE, write negative-loss bit patterns (sentinel 0xFFFFFFFF otherwise),
// per-row pos/neg counts, per-block positive-loss partial sums, level-0 histogram.
__global__ __launch_bounds__(TPB) void ohem_pass1_kernel(
    const float* __restrict__ logits, const int* __restrict__ targets,
    const int* __restrict__ mask, uint32_t* __restrict__ patterns,
    float* __restrict__ partials, int* __restrict__ nstats,
    uint32_t* __restrict__ hist0) {
    const int b = blockIdx.y;
    __shared__ unsigned int shist[2048];
    __shared__ float ssum[TPB];
    __shared__ int snp, snn;
    for (int i = threadIdx.x; i < 2048; i += TPB) shist[i] = 0u;
    if (threadIdx.x == 0) { snp = 0; snn = 0; }
    __syncthreads();

    const float4* lg4 = (const float4*)(logits  + (size_t)b * NPIX);
    const int4*   tg4 = (const int4*)(targets   + (size_t)b * NPIX);
    const int4*   mk4 = (const int4*)(mask      + (size_t)b * NPIX);
    uint4*        pt4 = (uint4*)(patterns       + (size_t)b * NPIX);

    const int nvec = NPIX / 4;
    const int stride = NPB * TPB;
    float sum_pos = 0.f;
    int np = 0, nn = 0;

    for (int i = blockIdx.x * TPB + threadIdx.x; i < nvec; i += stride) {
        __builtin_prefetch(lg4 + i + stride, 0, 0);   // -> global_prefetch_b8
        __builtin_prefetch(tg4 + i + stride, 0, 0);
        __builtin_prefetch(mk4 + i + stride, 0, 0);
        float4 x = lg4[i]; int4 t = tg4[i]; int4 m = mk4[i];
        uint4 pv;
#define OHEM_PROC(XC, TC, MC, PC)                                               \
        {                                                                       \
            float xx = (XC);                                                    \
            float l  = fmaxf(xx, 0.f) - xx * (float)(TC)                        \
                     + log1pf(expf(-fabsf(xx)));                                \
            uint32_t pp = 0xFFFFFFFFu;                                          \
            if (MC) {                                                           \
                if (TC) { np++; sum_pos += l; }                                 \
                else    { nn++; pp = __float_as_uint(l);                        \
                          atomicAdd(&shist[pp >> 20], 1u); }                    \
            }                                                                   \
            (PC) = pp;                                                          \
        }
        OHEM_PROC(x.x, t.x, m.x, pv.x)
        OHEM_PROC(x.y, t.y, m.y, pv.y)
        OHEM_PROC(x.z, t.z, m.z, pv.z)
        OHEM_PROC(x.w, t.w, m.w, pv.w)
#undef OHEM_PROC
        pt4[i] = pv;
    }

    atomicAdd(&snp, np);
    atomicAdd(&snn, nn);
    __syncthreads();
    for (int i = threadIdx.x; i < 2048; i += TPB) {
        unsigned int c = shist[i];
        if (c) atomicAdd(&hist0[(size_t)b * 2048 + i], c);
    }
    if (threadIdx.x == 0) {
        atomicAdd(&nstats[2 * b + 0], snp);
        atomicAdd(&nstats[2 * b + 1], snn);
    }
    // deterministic block-tree reduction of positive-loss sum
    ssum[threadIdx.x] = sum_pos;
    __syncthreads();
    for (int k = TPB / 2; k > 0; k >>= 1) {
        if (threadIdx.x < k) ssum[threadIdx.x] += ssum[threadIdx.x + k];
        __syncthreads();
    }
    if (threadIdx.x == 0) partials[(size_t)b * NPB + blockIdx.x] = ssum[0];
}

// ------------------------------------------------------ WMMA row reduce ---
// One wave per row. Uses V_WMMA_F32_16X16X4_F32 with B = ones as a 64-wide
// f32 reduction engine: sum over all C elements = 16 * sum of A inputs.
__global__ __launch_bounds__(32) void ohem_reduce_rows_kernel(
    const float* __restrict__ partials, float* __restrict__ out, int count) {
    const int b = blockIdx.x;
    const int lane = threadIdx.x;
    const float* p = partials + (size_t)b * count;
    float s;
#if __has_builtin(__builtin_amdgcn_wmma_f32_16x16x4_f32)
    v8f c = {0.f, 0.f, 0.f, 0.f, 0.f, 0.f, 0.f, 0.f};
    v2f onesv; onesv.x = 1.f; onesv.y = 1.f;
    int i = 0;
    for (; i + 64 <= count; i += 64) {
        v2f a;
        a.x = p[i + lane];
        a.y = p[i + 32 + lane];
        c = __builtin_amdgcn_wmma_f32_16x16x4_f32(false, a, false, onesv,
                                                  (short)0, c, false, false);
    }
    s = 0.f;
#pragma unroll
    for (int v = 0; v < 8; ++v) s += c[v];
    for (int j = i + lane; j < count; j += 32) s += 16.0f * p[j];
    for (int off = 16; off > 0; off >>= 1) s += __shfl_down(s, off);
    s *= (1.0f / 16.0f);
#else
    s = 0.f;
    for (int j = lane; j < count; j += 32) s += p[j];
    for (int off = 16; off > 0; off >>= 1) s += __shfl_down(s, off);
#endif
    if (lane == 0) out[b] = s;
}

// ----------------------------------------------------------- histogram ---
__global__ __launch_bounds__(TPB) void ohem_hist_kernel(
    const uint32_t* __restrict__ patterns, const uint32_t* __restrict__ prefix,
    const int* __restrict__ active, uint32_t* __restrict__ hist,
    int shift, int nbins, uint32_t himask) {
    const int b = blockIdx.y;
    if (!active[b]) return;
    __shared__ unsigned int shist[2048];
    for (int i = threadIdx.x; i < nbins; i += TPB) shist[i] = 0u;
    __syncthreads();
    const uint32_t want = prefix[b] & himask;
    const uint32_t binmask = (uint32_t)(nbins - 1);
    const uint4* pt4 = (const uint4*)(patterns + (size_t)b * NPIX);
    const int nvec = NPIX / 4;
    const int stride = HPB * TPB;
    for (int i = blockIdx.x * TPB + threadIdx.x; i < nvec; i += stride) {
        __builtin_prefetch(pt4 + i + stride, 0, 0);
        uint4 p = pt4[i];
        if (p.x < 0x80000000u && (p.x & himask) == want) atomicAdd(&shist[(p.x >> shift) & binmask], 1u);
        if (p.y < 0x80000000u && (p.y & himask) == want) atomicAdd(&shist[(p.y >> shift) & binmask], 1u);
        if (p.z < 0x80000000u && (p.z & himask) == want) atomicAdd(&shist[(p.z >> shift) & binmask], 1u);
        if (p.w < 0x80000000u && (p.w & himask) == want) atomicAdd(&shist[(p.w >> shift) & binmask], 1u);
    }
    __syncthreads();
    for (int i = threadIdx.x; i < nbins; i += TPB) {
        unsigned int c = shist[i];
        if (c) atomicAdd(&hist[(size_t)b * 2048 + i], c);
    }
}

// -------------------------------------------------------------- select ---
__global__ void ohem_select_kernel(const uint32_t* __restrict__ hist,
    const int* __restrict__ nstats, uint32_t* __restrict__ prefix,
    int* __restrict__ remaining, int* __restrict__ active,
    int* __restrict__ kkeep, int level, int shift, int nbins) {
    int b = threadIdx.x;
    if (b >= BATCH) return;
    if (level == 0) {
        int np = nstats[2 * b], nn = nstats[2 * b + 1];
        int m = KALL - np;
        if (m < 0) m = 0;
        if (m > nn) m = nn;
        kkeep[b] = np + m;
        if (m <= 0) { active[b] = 0; prefix[b] = 0xFFFFFFFFu; remaining[b] = 0; return; }
        active[b] = 1; prefix[b] = 0u; remaining[b] = m;
    }
    if (!active[b]) return;
    int rem = remaining[b];
    uint32_t pref = prefix[b];
    long long cum = 0;
    const uint32_t* h = hist + (size_t)b * 2048;
    for (int bin = nbins - 1; bin >= 0; --bin) {
        unsigned int cv = h[bin];
        if (cum + (long long)cv >= (long long)rem) {
            pref |= ((uint32_t)bin) << shift;
            rem -= (int)cum;
            prefix[b] = pref;
            remaining[b] = rem;
            return;
        }
        cum += cv;
    }
    prefix[b] = pref;      // unreachable if counts are consistent
    remaining[b] = 0;
}

// ------------------------------------------------------------ top sum  ---
__global__ __launch_bounds__(TPB) void ohem_sum_kernel(
    const uint32_t* __restrict__ patterns, const uint32_t* __restrict__ prefix,
    float* __restrict__ partials2) {
    const int b = blockIdx.y;
    __shared__ float ssum[TPB];
    const uint32_t T = prefix[b];          // 0xFFFFFFFF when inactive -> no hits
    const uint4* pt4 = (const uint4*)(patterns + (size_t)b * NPIX);
    const int nvec = NPIX / 4;
    const int stride = SPB * TPB;
    float s = 0.f;
    for (int i = blockIdx.x * TPB + threadIdx.x; i < nvec; i += stride) {
        __builtin_prefetch(pt4 + i + stride, 0, 0);
        uint4 p = pt4[i];
        if (p.x < 0x80000000u && p.x > T) s += __uint_as_float(p.x);
        if (p.y < 0x80000000u && p.y > T) s += __uint_as_float(p.y);
        if (p.z < 0x80000000u && p.z > T) s += __uint_as_float(p.z);
        if (p.w < 0x80000000u && p.w > T) s += __uint_as_float(p.w);
    }
    ssum[threadIdx.x] = s;
    __syncthreads();
    for (int k = TPB / 2; k > 0; k >>= 1) {
        if (threadIdx.x < k) ssum[threadIdx.x] += ssum[threadIdx.x + k];
        __syncthreads();
    }
    if (threadIdx.x == 0) partials2[(size_t)b * SPB + blockIdx.x] = ssum[0];
}

// ------------------------------------------------------------ finalize ---
__global__ void ohem_finalize_kernel(const float* __restrict__ sumPos,
    const float* __restrict__ sumG, const uint32_t* __restrict__ prefix,
    const int* __restrict__ remaining, const int* __restrict__ active,
    const int* __restrict__ kkeep, float* __restrict__ out) {
    if (threadIdx.x == 0 && blockIdx.x == 0) {
        float acc = 0.f;
        for (int b = 0; b < BATCH; ++b) {
            float st = sumG[b];
            if (active[b]) st += (float)remaining[b] * __uint_as_float(prefix[b]);
            int kk = kkeep[b];
            float per = (kk > 0) ? (sumPos[b] + st) / (float)kk : 0.f;
            acc += per;
        }
        out[0] = acc / (float)BATCH;
    }
}

// ---------------------------------------------------------------------------
extern "C" void kernel_launch(void* const* d_in, const int* in_sizes, int n_in,
                              void* d_out, int out_size, void* d_ws, size_t ws_size,
                              hipStream_t stream) {
    const float* logits  = (const float*)d_in[0];
    const int*   targets = (const int*)d_in[1];
    const int*   tissue  = (const int*)d_in[2];
    float* out = (float*)d_out;

    char* ws = (char*)d_ws;
    uint32_t* patterns  = (uint32_t*)(ws);                         // 33,554,432 B
    float*    partials1 = (float*)(ws + 33554432);                 //     16,384 B
    float*    partials2 = (float*)(ws + 33570816);                 //     16,384 B
    uint32_t* hist      = (uint32_t*)(ws + 33587200);              //    196,608 B
    int*      nstats    = (int*)(ws + 33783808);                   //         64 B
    uint32_t* prefix    = (uint32_t*)(ws + 33783872);              //         32 B
    int*      remaining = (int*)(ws + 33783904);                   //         32 B
    int*      active    = (int*)(ws + 33783936);                   //         32 B
    int*      kkeep     = (int*)(ws + 33783968);                   //         32 B
    float*    sumPos    = (float*)(ws + 33784000);                 //         32 B
    float*    sumG      = (float*)(ws + 33784032);                 //         32 B

    ohem_init_kernel<<<dim3((3 * BATCH * 2048 + 255) / 256), 256, 0, stream>>>(hist, nstats);

    ohem_pass1_kernel<<<dim3(NPB, BATCH), TPB, 0, stream>>>(
        logits, targets, tissue, patterns, partials1, nstats, hist);

    ohem_reduce_rows_kernel<<<BATCH, 32, 0, stream>>>(partials1, sumPos, NPB);

    // level 0: bits [30:20]
    ohem_select_kernel<<<1, 32, 0, stream>>>(hist, nstats, prefix, remaining,
                                             active, kkeep, 0, 20, 2048);
    // level 1: bits [19:9]
    ohem_hist_kernel<<<dim3(HPB, BATCH), TPB, 0, stream>>>(
        patterns, prefix, active, hist + (size_t)BATCH * 2048, 9, 2048, 0x7FF00000u);
    ohem_select_kernel<<<1, 32, 0, stream>>>(hist + (size_t)BATCH * 2048, nstats,
                                             prefix, remaining, active, kkeep, 1, 9, 2048);
    // level 2: bits [8:0]
    ohem_hist_kernel<<<dim3(HPB, BATCH), TPB, 0, stream>>>(
        patterns, prefix, active, hist + (size_t)2 * BATCH * 2048, 0, 512, 0x7FFFFE00u);
    ohem_select_kernel<<<1, 32, 0, stream>>>(hist + (size_t)2 * BATCH * 2048, nstats,
                                             prefix, remaining, active, kkeep, 2, 0, 512);

    ohem_sum_kernel<<<dim3(SPB, BATCH), TPB, 0, stream>>>(patterns, prefix, partials2);
    ohem_reduce_rows_kernel<<<BATCH, 32, 0, stream>>>(partials2, sumG, SPB);

    ohem_finalize_kernel<<<1, 32, 0, stream>>>(sumPos, sumG, prefix, remaining,
                                               active, kkeep, out);
}